// TrafficGATv2_56057913147872
// MI455X (gfx1250) — compile-verified
//
#include <hip/hip_runtime.h>
#include <math.h>

#define NN 50000
#define EEDGE 800000
#define HH 4
#define CC 64
#define HC 256            // HH*CC
#define EA 850000         // EEDGE + NN (self-loop augmented)
#define EATILES (EA / 16) // 53125

typedef float v2f __attribute__((ext_vector_type(2)));
typedef float v8f __attribute__((ext_vector_type(8)));

// ---------- float atomic max (amax initialized to -inf) ----------
__device__ __forceinline__ void atomicMaxF(float* addr, float val) {
  if (val >= 0.0f)
    atomicMax((int*)addr, __float_as_int(val));
  else
    atomicMin((unsigned int*)addr, (unsigned int)__float_as_int(val));
}

// ---------- encoders ----------
__global__ void encode_nodes_k(const float* __restrict__ x, const float* __restrict__ w,
                               const float* __restrict__ b, float* __restrict__ h) {
  int idx = blockIdx.x * blockDim.x + threadIdx.x;
  if (idx >= NN * CC) return;
  int n = idx >> 6, c = idx & 63;
  float v = b[c];
#pragma unroll
  for (int k = 0; k < 4; ++k) v += x[n * 4 + k] * w[k * CC + c];
  h[idx] = v;
}

__global__ void encode_edges_k(const float* __restrict__ ea, const float* __restrict__ w,
                               const float* __restrict__ b, float* __restrict__ out) {
  int idx = blockIdx.x * blockDim.x + threadIdx.x;
  if (idx >= EEDGE * CC) return;
  int e = idx >> 6, c = idx & 63;
  float v = b[c];
#pragma unroll
  for (int k = 0; k < 3; ++k) v += ea[e * 3 + k] * w[k * CC + c];
  out[idx] = v;
}

// ---------- self-loop edge attr = mean of incoming edge embeddings ----------
__global__ void loop_accum_k(const float* __restrict__ ee, const int* __restrict__ ei,
                             float* __restrict__ lsum, float* __restrict__ cnt) {
  int idx = blockIdx.x * blockDim.x + threadIdx.x;
  if (idx >= EEDGE * CC) return;
  int e = idx >> 6, c = idx & 63;
  int d = ei[EEDGE + e];
  atomicAdd(&lsum[d * CC + c], ee[idx]);
  if (c == 0) atomicAdd(&cnt[d], 1.0f);
}

__global__ void loop_final_k(float* __restrict__ lsum, const float* __restrict__ cnt) {
  int idx = blockIdx.x * blockDim.x + threadIdx.x;
  if (idx >= NN * CC) return;
  lsum[idx] /= fmaxf(cnt[idx >> 6], 1.0f);
}

// ---------- WMMA fp32 GEMM: out[M,256] = A[M,64] @ W[64,256] + bias ----------
// One wave per 16x16 output tile; 16 steps of V_WMMA_F32_16X16X4_F32 (K=64).
__global__ void __launch_bounds__(32) gemm_n64_k(const float* __restrict__ A,
                                                 const float* __restrict__ W,
                                                 const float* __restrict__ bias,
                                                 float* __restrict__ out) {
  int lane = threadIdx.x;
  int half = lane >> 4, ln = lane & 15;
  int r0 = blockIdx.x * 16;
  int colb = blockIdx.y * 16;
  const float* arow = A + (size_t)(r0 + ln) * CC;
  v8f acc = {};
#pragma unroll
  for (int kk = 0; kk < 16; ++kk) {
    int kb = kk * 4 + 2 * half;              // A/B frag: K = vgpr_j + 2*(lane>>4)
    v2f a;
    a.x = arow[kb];
    a.y = arow[kb + 1];
    const float* wp = W + (size_t)kb * HC + colb + ln;
    v2f b;
    b.x = wp[0];
    b.y = wp[HC];
    acc = __builtin_amdgcn_wmma_f32_16x16x4_f32(false, a, false, b, (short)0, acc,
                                                false, false);
  }
  int col = colb + ln;
  float bv = bias[col];
#pragma unroll
  for (int r = 0; r < 8; ++r)
    out[(size_t)(r0 + r + 8 * half) * HC + col] = acc[r] + bv;
}

// ---------- per-layer init: acc=0, amax=-inf, denom=0 ----------
__global__ void init_layer_k(float* __restrict__ acc, float* __restrict__ amax,
                             float* __restrict__ denom) {
  int idx = blockIdx.x * blockDim.x + threadIdx.x;
  if (idx < NN * HC) acc[idx] = 0.0f;
  if (idx < NN * HH) {
    amax[idx] = -INFINITY;
    denom[idx] = 0.0f;
  }
}

// ---------- pass 1: fused ye-GEMM (WMMA) + leaky-relu + attention logits ----------
// 8 waves/block, one 16-edge tile per wave. ee_a tile staged in LDS; We streamed
// from L2. Logits reduced with a 16-lane shuffle butterfly; amax via atomic max.
__global__ void __launch_bounds__(256) attn_logits_k(
    const float* __restrict__ ee, const float* __restrict__ loopa,
    const float* __restrict__ xl, const float* __restrict__ xr,
    const float* __restrict__ We, const float* __restrict__ att,
    const int* __restrict__ ei, float* __restrict__ logits,
    float* __restrict__ amax) {
  __shared__ float sA[8][16 * CC];
  __shared__ int sIdx[8][32];
  int w = threadIdx.x >> 5;
  int lane = threadIdx.x & 31;
  int tile = blockIdx.x * 8 + w;
  if (tile >= EATILES) return;   // wave-uniform exit; no cross-wave syncs below
  int r0 = tile * 16;

  // stage the wave's 16 augmented-edge feature rows (16x64 fp32) into LDS
  for (int i = 0; i < 32; ++i) {
    int idx = i * 32 + lane;
    int row = idx >> 6, k = idx & 63;
    int e = r0 + row;
    sA[w][idx] = (e < EEDGE) ? ee[(size_t)e * CC + k]
                             : loopa[(size_t)(e - EEDGE) * CC + k];
  }
  // stage src (lanes 0-15) / dst (lanes 16-31) of the 16 edges
  {
    int row = lane & 15;
    int e = r0 + row;
    int v;
    if (lane < 16)
      v = (e < EEDGE) ? ei[e] : (e - EEDGE);
    else
      v = (e < EEDGE) ? ei[EEDGE + e] : (e - EEDGE);
    sIdx[w][lane] = v;
  }
  // per-wave LDS write->read: DS ops are in-order within a wave (DScnt)

  int half = lane >> 4, ln = lane & 15;
  for (int head = 0; head < HH; ++head) {
    float part[8];
#pragma unroll
    for (int r = 0; r < 8; ++r) part[r] = 0.0f;
    for (int sub = 0; sub < 4; ++sub) {
      int colb = (head * 4 + sub) * 16;
      v8f acc = {};
#pragma unroll
      for (int kk = 0; kk < 16; ++kk) {
        int kb = kk * 4 + 2 * half;
        v2f a;
        a.x = sA[w][ln * CC + kb];
        a.y = sA[w][ln * CC + kb + 1];
        const float* wp = We + (size_t)kb * HC + colb + ln;
        v2f b;
        b.x = wp[0];
        b.y = wp[HC];
        acc = __builtin_amdgcn_wmma_f32_16x16x4_f32(false, a, false, b, (short)0,
                                                    acc, false, false);
      }
      int col = colb + ln;
      float av = att[head * CC + (col & 63)];
#pragma unroll
      for (int r = 0; r < 8; ++r) {
        int row = r + 8 * half;              // C/D layout: M = r + 8*(lane>>4)
        int s = sIdx[w][row];
        int d = sIdx[w][16 + row];
        float m = acc[r] + xl[(size_t)s * HC + col] + xr[(size_t)d * HC + col];
        m = (m > 0.0f) ? m : 0.2f * m;       // leaky_relu(0.2)
        part[r] += m * av;
      }
    }
    // sum over the 16 lanes of each half (xor 1,2,4,8 stays inside a half)
#pragma unroll
    for (int off = 1; off < 16; off <<= 1) {
#pragma unroll
      for (int r = 0; r < 8; ++r) part[r] += __shfl_xor(part[r], off, 32);
    }
    if (ln == 0) {
#pragma unroll
      for (int r = 0; r < 8; ++r) {
        int row = r + 8 * half;
        int e = r0 + row;
        logits[(size_t)e * HH + head] = part[r];
        atomicMaxF(&amax[sIdx[w][16 + row] * HH + head], part[r]);
      }
    }
  }
}

// ---------- pass 2: unnormalized softmax scatter-sum into L2-resident acc ----------
__global__ void attn_accum_k(const float* __restrict__ logits,
                             const float* __restrict__ amax,
                             const float* __restrict__ xl, const int* __restrict__ ei,
                             float* __restrict__ denom, float* __restrict__ acc) {
  size_t idx = (size_t)blockIdx.x * blockDim.x + threadIdx.x;
  int e = (int)(idx >> 8);
  int hc = (int)(idx & 255);
  if (e >= EA) return;
  int h = hc >> 6;
  int s, d;
  if (e < EEDGE) {
    s = ei[e];
    d = ei[EEDGE + e];
  } else {
    s = e - EEDGE;
    d = s;
  }
  float a = expf(logits[(size_t)e * HH + h] - amax[d * HH + h]);
  atomicAdd(&acc[(size_t)d * HC + hc], a * xl[(size_t)s * HC + hc]);
  if ((hc & 63) == 0) atomicAdd(&denom[d * HH + h], a);
}

// ---------- pass 3: normalize, mean over heads, +cbias, optional ELU ----------
__global__ void node_out_k(const float* __restrict__ acc, const float* __restrict__ denom,
                           const float* __restrict__ cbias, float* __restrict__ hout,
                           int do_elu) {
  int idx = blockIdx.x * blockDim.x + threadIdx.x;
  if (idx >= NN * CC) return;
  int n = idx >> 6, c = idx & 63;
  float v = 0.0f;
#pragma unroll
  for (int h = 0; h < HH; ++h)
    v += acc[(size_t)n * HC + h * CC + c] / (denom[n * HH + h] + 1e-16f);
  v = v * 0.25f + cbias[c];
  if (do_elu) v = (v > 0.0f) ? v : expm1f(v);
  hout[idx] = v;
}

// ---------- predictor MLP via WMMA ----------
// One wave per 16-edge tile: ctx[16,192] = [h[src] | h[dst] | ee] staged in LDS,
// hid[16,64] = relu(ctx @ pw1 + pb1) via 48 K-steps x 4 N-tiles of f32 WMMA,
// out[16] = hid @ pw2 + pb2 via per-lane dot + 16-lane shuffle reduction.
// 4 waves/block -> 48KB LDS; 50000 tiles / 4 = 12500 blocks exactly (no guards).
__global__ void __launch_bounds__(128) predictor_k(
    const float* __restrict__ h, const float* __restrict__ ee,
    const int* __restrict__ ei, const float* __restrict__ pw1,
    const float* __restrict__ pb1, const float* __restrict__ pw2,
    const float* __restrict__ pb2, float* __restrict__ out) {
  __shared__ float sC[4][16 * 192];
  int w = threadIdx.x >> 5;
  int lane = threadIdx.x & 31;
  int tile = blockIdx.x * 4 + w;
  int r0 = tile * 16;
  int half = lane >> 4, ln = lane & 15;

  // stage ctx rows: lane handles row (lane&15), k-range half*96..half*96+95
  {
    int row = ln;
    int e = r0 + row;
    int s = ei[e];
    int d = ei[EEDGE + e];
    int kbase = half * 96;
    for (int i = 0; i < 96; ++i) {
      int k = kbase + i;
      float v;
      if (k < 64)
        v = h[(size_t)s * CC + k];
      else if (k < 128)
        v = h[(size_t)d * CC + (k - 64)];
      else
        v = ee[(size_t)e * CC + (k - 128)];
      sC[w][row * 192 + k] = v;
    }
  }
  // per-wave LDS write->read: in-order DS within a wave, no block sync needed

  float part[8];
#pragma unroll
  for (int r = 0; r < 8; ++r) part[r] = 0.0f;
#pragma unroll
  for (int ct = 0; ct < 4; ++ct) {
    int colb = ct * 16;
    v8f acc = {};
#pragma unroll 8
    for (int kk = 0; kk < 48; ++kk) {
      int kb = kk * 4 + 2 * half;
      v2f a;
      a.x = sC[w][ln * 192 + kb];
      a.y = sC[w][ln * 192 + kb + 1];
      const float* wp = pw1 + (size_t)kb * 64 + colb + ln;
      v2f b;
      b.x = wp[0];
      b.y = wp[64];
      acc = __builtin_amdgcn_wmma_f32_16x16x4_f32(false, a, false, b, (short)0,
                                                  acc, false, false);
    }
    int col = colb + ln;
    float bb = pb1[col];
    float wv = pw2[col];
#pragma unroll
    for (int r = 0; r < 8; ++r) {
      float hid = fmaxf(acc[r] + bb, 0.0f);
      part[r] += hid * wv;
    }
  }
#pragma unroll
  for (int off = 1; off < 16; off <<= 1) {
#pragma unroll
    for (int r = 0; r < 8; ++r) part[r] += __shfl_xor(part[r], off, 32);
  }
  if (ln == 0) {
#pragma unroll
    for (int r = 0; r < 8; ++r)
      out[r0 + r + 8 * half] = part[r] + pb2[0];
  }
}

extern "C" void kernel_launch(void* const* d_in, const int* in_sizes, int n_in,
                              void* d_out, int out_size, void* d_ws, size_t ws_size,
                              hipStream_t stream) {
  const float* x      = (const float*)d_in[0];
  const float* eattr  = (const float*)d_in[1];
  const float* enc_w  = (const float*)d_in[2];
  const float* enc_b  = (const float*)d_in[3];
  const float* eenc_w = (const float*)d_in[4];
  const float* eenc_b = (const float*)d_in[5];
  const float* Wl     = (const float*)d_in[6];
  const float* bl     = (const float*)d_in[7];
  const float* Wr     = (const float*)d_in[8];
  const float* br     = (const float*)d_in[9];
  const float* We     = (const float*)d_in[10];
  const float* att    = (const float*)d_in[11];
  const float* cbias  = (const float*)d_in[12];
  const float* pw1    = (const float*)d_in[13];
  const float* pb1    = (const float*)d_in[14];
  const float* pw2    = (const float*)d_in[15];
  const float* pb2    = (const float*)d_in[16];
  const int*   ei     = (const int*)d_in[17];
  float* out = (float*)d_out;

  // workspace layout (~412 MB of fp32)
  float* w = (float*)d_ws;
  size_t o = 0;
  float* hA  = w + o; o += (size_t)NN * CC;
  float* hB  = w + o; o += (size_t)NN * CC;
  float* ee  = w + o; o += (size_t)EEDGE * CC;
  float* la  = w + o; o += (size_t)NN * CC;   // self-loop attr (sum -> mean)
  float* cnt = w + o; o += (size_t)NN;
  float* xl  = w + o; o += (size_t)NN * HC;
  float* xr  = w + o; o += (size_t)NN * HC;
  float* lg  = w + o; o += (size_t)EA * HH;   // attention logits
  float* am  = w + o; o += (size_t)NN * HH;   // segment max
  float* dn  = w + o; o += (size_t)NN * HH;   // softmax denom
  float* ac  = w + o; o += (size_t)NN * HC;   // weighted scatter-sum accumulator

  encode_nodes_k<<<(NN * CC + 255) / 256, 256, 0, stream>>>(x, enc_w, enc_b, hA);
  encode_edges_k<<<(EEDGE * CC + 255) / 256, 256, 0, stream>>>(eattr, eenc_w, eenc_b, ee);
  hipMemsetAsync(la, 0, (size_t)NN * CC * sizeof(float), stream);
  hipMemsetAsync(cnt, 0, (size_t)NN * sizeof(float), stream);
  loop_accum_k<<<(EEDGE * CC + 255) / 256, 256, 0, stream>>>(ee, ei, la, cnt);
  loop_final_k<<<(NN * CC + 255) / 256, 256, 0, stream>>>(la, cnt);

  float* hin = hA;
  float* hout = hB;
  for (int i = 0; i < 3; ++i) {
    const float* Wl_i = Wl + (size_t)i * CC * HC;
    const float* Wr_i = Wr + (size_t)i * CC * HC;
    const float* We_i = We + (size_t)i * CC * HC;
    gemm_n64_k<<<dim3(NN / 16, HC / 16), 32, 0, stream>>>(hin, Wl_i, bl + (size_t)i * HC, xl);
    gemm_n64_k<<<dim3(NN / 16, HC / 16), 32, 0, stream>>>(hin, Wr_i, br + (size_t)i * HC, xr);
    init_layer_k<<<(NN * HC + 255) / 256, 256, 0, stream>>>(ac, am, dn);
    attn_logits_k<<<(EATILES + 7) / 8, 256, 0, stream>>>(
        ee, la, xl, xr, We_i, att + (size_t)i * HH * CC, ei, lg, am);
    attn_accum_k<<<EA, 256, 0, stream>>>(lg, am, xl, ei, dn, ac);
    node_out_k<<<(NN * CC + 255) / 256, 256, 0, stream>>>(
        ac, dn, cbias + (size_t)i * CC, hout, i < 2 ? 1 : 0);
    float* tmp = hin; hin = hout; hout = tmp;
  }
  // hin now points at the layer-2 output
  predictor_k<<<EEDGE / 16 / 4, 128, 0, stream>>>(hin, ee, ei, pw1, pb1, pw2, pb2, out);
}